// LSTM_cell_62612033241410
// MI455X (gfx1250) — compile-verified
//
#include <hip/hip_runtime.h>
#include <hip/hip_bf16.h>
#include <math.h>

typedef float v2f __attribute__((ext_vector_type(2)));
typedef float v8f __attribute__((ext_vector_type(8)));

#define B_  256
#define O_  128
#define H_  1024
#define Hh_ 512
#define I_  2048
#define DF_ 1536

#define KT 32          // K-chunk (floats)
#define LS 36          // LDS row stride: 36 = 4*9 (odd) mod 64 -> conflict-free frags

__device__ __forceinline__ v8f wmma4(v2f a, v2f b, v8f c) {
    return __builtin_amdgcn_wmma_f32_16x16x4_f32(false, a, false, b, (short)0, c, false, false);
}
__device__ __forceinline__ float sigmoidf_(float x) { return 1.0f / (1.0f + __expf(-x)); }

// Direct (small) fragment load: lane L -> row (base + L&15), K-pair (k + 2*(L>>4)).
__device__ __forceinline__ v2f load_frag_g(const float* __restrict__ p, int rowBase,
                                           int ld, int kBase, int lane) {
    return *(const v2f*)(p + (size_t)(rowBase + (lane & 15)) * ld + kBase + ((lane >> 4) << 1));
}

// ---------------- K1: e_pre = h @ W_e^T + b_e  (256x512, K=1024; ~1% of work) --
__global__ __launch_bounds__(256) void k1_epre(const float* __restrict__ h,
                                               const float* __restrict__ We,
                                               const float* __restrict__ be,
                                               float* __restrict__ epre) {
    const int lane = threadIdx.x & 31;
    const int wave = threadIdx.x >> 5;     // 8 waves cover N=512
    const int mBase = blockIdx.x * 16;     // 16 blocks over M=256
    const int nBase = wave * 64;
    v8f acc[4] = {};
    for (int k = 0; k < H_; k += 4) {
        v2f a = load_frag_g(h, mBase, H_, k, lane);
#pragma unroll
        for (int j = 0; j < 4; ++j)
            acc[j] = wmma4(a, load_frag_g(We, nBase + j * 16, H_, k, lane), acc[j]);
    }
    const int mOff = 8 * (lane >> 4), nLoc = lane & 15;
#pragma unroll
    for (int j = 0; j < 4; ++j) {
        const int n = nBase + j * 16 + nLoc;
        const float bias = be[n];
#pragma unroll
        for (int r = 0; r < 8; ++r)
            epre[(size_t)(mBase + mOff + r) * Hh_ + n] = acc[j][r] + bias;
    }
}

// ---------------- K2: partial e[m] += sum_n w[n]*tanh(epre + x_j@U_e^T) -------
// GEMM M=32768, N=512, K=512. Block tile 128M x 128N, 8 waves of 32M x 64N.
// LDS-staged, double-buffered, coalesced global loads, conflict-free frags.
__global__ __launch_bounds__(256) void k2_escore(const float* __restrict__ xj,
                                                 const float* __restrict__ Ue,
                                                 const float* __restrict__ wv,
                                                 const float* __restrict__ epre,
                                                 float* __restrict__ e) {
    __shared__ float lA[2][128 * LS];
    __shared__ float lB[2][128 * LS];
    __shared__ float part[8][32];

    const int t = threadIdx.x;
    const int lane = t & 31;
    const int wave = t >> 5;
    const int wm = wave >> 1;            // 0..3 : 32-row strip
    const int wn = wave & 1;             // 0..1 : 64-col strip
    const int mBase = blockIdx.x * 128;  // one batch row b == blockIdx.x (O_==128)
    const int nBlk  = blockIdx.y * 128;
    const int half = lane >> 4, lr = lane & 15;

    float4 rg[8];
    auto gfetch = [&](int kc) {
#pragma unroll
        for (int p = 0; p < 8; ++p) {
            const int ei = p * 256 + t;
            const int row = ei >> 3, q = ei & 7;       // 256 rows x 8 quads
            const float* src = (row < 128)
                ? xj + (size_t)(mBase + row) * Hh_ + kc + q * 4
                : Ue + (size_t)(nBlk + row - 128) * Hh_ + kc + q * 4;
            rg[p] = *(const float4*)src;
        }
    };
    auto lstore = [&](int buf) {
#pragma unroll
        for (int p = 0; p < 8; ++p) {
            const int ei = p * 256 + t;
            const int row = ei >> 3, q = ei & 7;
            float* dst = (row < 128) ? &lA[buf][row * LS + q * 4]
                                     : &lB[buf][(row - 128) * LS + q * 4];
            *(float4*)dst = rg[p];
        }
    };

    v8f acc[2][4] = {};
    gfetch(0);
    lstore(0);
    __syncthreads();
    const int NC = Hh_ / KT;   // 16 chunks
    for (int cc = 0; cc < NC; ++cc) {
        if (cc + 1 < NC) gfetch((cc + 1) * KT);
        const float* A  = lA[cc & 1];
        const float* Bm = lB[cc & 1];
#pragma unroll
        for (int ks = 0; ks < KT / 4; ++ks) {
            v2f af[2], bf[4];
#pragma unroll
            for (int i = 0; i < 2; ++i)
                af[i] = *(const v2f*)&A[(wm * 32 + i * 16 + lr) * LS + ks * 4 + 2 * half];
#pragma unroll
            for (int j = 0; j < 4; ++j)
                bf[j] = *(const v2f*)&Bm[(wn * 64 + j * 16 + lr) * LS + ks * 4 + 2 * half];
#pragma unroll
            for (int i = 0; i < 2; ++i)
#pragma unroll
                for (int j = 0; j < 4; ++j)
                    acc[i][j] = wmma4(af[i], bf[j], acc[i][j]);
        }
        if (cc + 1 < NC) lstore((cc + 1) & 1);
        __syncthreads();
    }

    // fused epilogue: tanh, dot with w, reduce, accumulate partial into e[m]
    const float* epreRow = epre + (size_t)blockIdx.x * Hh_;
#pragma unroll
    for (int i = 0; i < 2; ++i)
#pragma unroll
        for (int r = 0; r < 8; ++r) {
            float s = 0.0f;
#pragma unroll
            for (int j = 0; j < 4; ++j) {
                const int n = nBlk + wn * 64 + j * 16 + lr;
                s += wv[n] * tanhf(acc[i][j][r] + epreRow[n]);
            }
#pragma unroll
            for (int off = 1; off < 16; off <<= 1) s += __shfl_xor(s, off, 32);
            if (lr == 0) part[wave][i * 16 + r + 8 * half] = s;
        }
    __syncthreads();
    if (t < 128) {
        const int wmq = t >> 5;
        const float s = part[wmq * 2 + 0][t & 31] + part[wmq * 2 + 1][t & 31];
        atomicAdd(&e[mBase + t], s);
    }
}

// ---------------- K3: masked softmax over O per batch row --------------------
__global__ __launch_bounds__(128) void k3_softmax(const float* __restrict__ e,
                                                  const unsigned char* __restrict__ mask,
                                                  float* __restrict__ a) {
    __shared__ float red[128];
    const int b = blockIdx.x, t = threadIdx.x;
    const float v = mask[(size_t)b * O_ + t] ? e[(size_t)b * O_ + t] : -1e9f;
    red[t] = v; __syncthreads();
    for (int s = 64; s > 0; s >>= 1) { if (t < s) red[t] = fmaxf(red[t], red[t + s]); __syncthreads(); }
    const float mx = red[0]; __syncthreads();
    const float ex = __expf(v - mx);
    red[t] = ex; __syncthreads();
    for (int s = 64; s > 0; s >>= 1) { if (t < s) red[t] += red[t + s]; __syncthreads(); }
    a[(size_t)b * O_ + t] = ex / red[0];
}

// ---------------- K4: x = concat(sum_o a[b,o]*x_j[b,o,:], X_F[b,:]) ----------
__global__ __launch_bounds__(256) void k4_phi_concat(const float* __restrict__ a,
                                                     const float* __restrict__ xj,
                                                     const float* __restrict__ XF,
                                                     float* __restrict__ x) {
    __shared__ float aS[O_];
    const int b = blockIdx.x, t = threadIdx.x;
    if (t < O_) aS[t] = a[(size_t)b * O_ + t];
    __syncthreads();
    const float* xjb = xj + (size_t)b * O_ * Hh_;
#pragma unroll
    for (int f0 = 0; f0 < Hh_; f0 += 256) {
        const int f = f0 + t;
        float acc = 0.0f;
#pragma unroll 4
        for (int o = 0; o < O_; ++o) acc += aS[o] * xjb[(size_t)o * Hh_ + f];
        x[(size_t)b * I_ + f] = acc;
    }
    for (int f = t; f < DF_; f += 256)
        x[(size_t)b * I_ + Hh_ + f] = XF[(size_t)b * DF_ + f];
}

// ---------------- K5: 4-gate GEMM pre-activations, K-split of 2 --------------
// Block tile 64M x 64N x 4 gates; 8 waves = (gate, m-half of 32 rows).
// ks=0: x@W (K=2048). ks=1: h@U then c@V (K=2048, V_c tile zero-filled).
__global__ __launch_bounds__(256) void k5_gates(const float* __restrict__ x,
                                                const float* __restrict__ h,
                                                const float* __restrict__ c,
                                                const float* __restrict__ W0, const float* __restrict__ W1,
                                                const float* __restrict__ W2, const float* __restrict__ W3,
                                                const float* __restrict__ U0, const float* __restrict__ U1,
                                                const float* __restrict__ U2, const float* __restrict__ U3,
                                                const float* __restrict__ V0, const float* __restrict__ V1,
                                                const float* __restrict__ V3,
                                                float* __restrict__ pre) {
    __shared__ float lA[2][64 * LS];
    __shared__ float lB[2][256 * LS];

    const int t = threadIdx.x;
    const int lane = t & 31;
    const int wave = t >> 5;
    const int g  = wave >> 1;            // gate 0=i 1=f 2=c 3=o
    const int ms = wave & 1;             // m-half (32 rows)
    const int mBase = blockIdx.x * 64;   // 4 tiles over M=256
    const int nBlk  = blockIdx.y * 64;   // 16 tiles over N=1024
    const int ksp   = blockIdx.z;        // K split: 0 or 1
    const int half = lane >> 4, lr = lane & 15;

    const float* Wg[4] = {W0, W1, W2, W3};
    const float* Ug[4] = {U0, U1, U2, U3};
    const float* Vg[4] = {V0, V1, (const float*)0, V3};

    float4 rg[10];
    auto gfetch = [&](int kc) {
#pragma unroll
        for (int p = 0; p < 10; ++p) {
            const int ei = p * 256 + t;
            const int row = ei >> 3, q = ei & 7;       // 320 rows x 8 quads
            const float* src;
            if (row < 64) {   // A rows
                src = (ksp == 0) ? x + (size_t)(mBase + row) * I_ + kc + q * 4
                    : (kc < 1024) ? h + (size_t)(mBase + row) * H_ + kc + q * 4
                                  : c + (size_t)(mBase + row) * H_ + (kc - 1024) + q * 4;
            } else {          // B rows: 4 gates x 64
                const int gg = (row - 64) >> 6, br = (row - 64) & 63;
                const int n = nBlk + br;
                if (ksp == 0)          src = Wg[gg] + (size_t)n * I_ + kc + q * 4;
                else if (kc < 1024)    src = Ug[gg] + (size_t)n * H_ + kc + q * 4;
                else                   src = Vg[gg] ? Vg[gg] + (size_t)n * H_ + (kc - 1024) + q * 4
                                                    : (const float*)0;
            }
            rg[p] = src ? *(const float4*)src : make_float4(0.f, 0.f, 0.f, 0.f);
        }
    };
    auto lstore = [&](int buf) {
#pragma unroll
        for (int p = 0; p < 10; ++p) {
            const int ei = p * 256 + t;
            const int row = ei >> 3, q = ei & 7;
            float* dst = (row < 64) ? &lA[buf][row * LS + q * 4]
                                    : &lB[buf][(row - 64) * LS + q * 4];
            *(float4*)dst = rg[p];
        }
    };

    v8f acc[2][4] = {};
    gfetch(0);
    lstore(0);
    __syncthreads();
    const int NC = 2048 / KT;    // 64 chunks
    for (int cc = 0; cc < NC; ++cc) {
        if (cc + 1 < NC) gfetch((cc + 1) * KT);
        const float* A  = lA[cc & 1];
        const float* Bm = lB[cc & 1];
#pragma unroll
        for (int ks = 0; ks < KT / 4; ++ks) {
            v2f af[2], bf[4];
#pragma unroll
            for (int i = 0; i < 2; ++i)
                af[i] = *(const v2f*)&A[(ms * 32 + i * 16 + lr) * LS + ks * 4 + 2 * half];
#pragma unroll
            for (int j = 0; j < 4; ++j)
                bf[j] = *(const v2f*)&Bm[(g * 64 + j * 16 + lr) * LS + ks * 4 + 2 * half];
#pragma unroll
            for (int i = 0; i < 2; ++i)
#pragma unroll
                for (int j = 0; j < 4; ++j)
                    acc[i][j] = wmma4(af[i], bf[j], acc[i][j]);
        }
        if (cc + 1 < NC) lstore((cc + 1) & 1);
        __syncthreads();
    }

    float* dst = pre + ((size_t)(ksp * 4 + g) << 18);   // B_*H_ = 1<<18
#pragma unroll
    for (int i = 0; i < 2; ++i)
#pragma unroll
        for (int j = 0; j < 4; ++j) {
            const int n = nBlk + j * 16 + lr;
#pragma unroll
            for (int r = 0; r < 8; ++r) {
                const int m = mBase + ms * 32 + i * 16 + r + 8 * half;
                dst[(size_t)m * H_ + n] = acc[i][j][r];
            }
        }
}

// ---------------- K6: elementwise peephole LSTM epilogue ---------------------
__global__ __launch_bounds__(256) void k6_lstm(const float* __restrict__ pre,
                                               const float* __restrict__ cin,
                                               const float* __restrict__ bi, const float* __restrict__ bf,
                                               const float* __restrict__ bc, const float* __restrict__ bo,
                                               float* __restrict__ out) {
    const int idx = blockIdx.x * 256 + threadIdx.x;   // over B_*H_
    const int n = idx & (H_ - 1);
    const int BH = B_ * H_;
    const float pi = pre[idx]          + pre[4 * BH + idx] + bi[n];
    const float pf = pre[BH + idx]     + pre[5 * BH + idx] + bf[n];
    const float pc = pre[2 * BH + idx] + pre[6 * BH + idx] + bc[n];
    const float po = pre[3 * BH + idx] + pre[7 * BH + idx] + bo[n];
    const float cv = cin[idx];
    const float iv = sigmoidf_(pi), fv = sigmoidf_(pf), ov = sigmoidf_(po);
    const float gv = tanhf(pc);
    const float nc = fv * cv + iv * gv;
    out[idx] = ov * tanhf(nc);          // new_h
    out[BH + idx] = nc;                 // new_c
}

extern "C" void kernel_launch(void* const* d_in, const int* in_sizes, int n_in,
                              void* d_out, int out_size, void* d_ws, size_t ws_size,
                              hipStream_t stream) {
    const float* xj   = (const float*)d_in[0];
    const unsigned char* mask = (const unsigned char*)d_in[1];
    const float* XF   = (const float*)d_in[2];
    const float* h    = (const float*)d_in[3];
    const float* c    = (const float*)d_in[4];
    const float* Wi   = (const float*)d_in[5];
    const float* Wf   = (const float*)d_in[6];
    const float* Wc   = (const float*)d_in[7];
    const float* Wo   = (const float*)d_in[8];
    const float* Ui   = (const float*)d_in[9];
    const float* Uf   = (const float*)d_in[10];
    const float* Uc   = (const float*)d_in[11];
    const float* Uo   = (const float*)d_in[12];
    const float* Vi   = (const float*)d_in[13];
    const float* Vf   = (const float*)d_in[14];
    /* V_c = d_in[15] unused by reference */
    const float* Vo   = (const float*)d_in[16];
    const float* bi   = (const float*)d_in[17];
    const float* bf   = (const float*)d_in[18];
    const float* bc   = (const float*)d_in[19];
    const float* bo   = (const float*)d_in[20];
    const float* wv   = (const float*)d_in[21];
    const float* We   = (const float*)d_in[22];
    const float* Ue   = (const float*)d_in[23];
    const float* be   = (const float*)d_in[24];

    float* out  = (float*)d_out;
    float* ws   = (float*)d_ws;
    float* epre = ws;                                  // 131072 f32
    float* e    = epre + 131072;                       //  32768 f32
    float* a    = e + 32768;                           //  32768 f32
    float* x    = a + 32768;                           // 524288 f32
    float* pre  = x + 524288;                          // 2 * 4 * 262144 f32 (~8 MB)

    k1_epre     <<<16, 256, 0, stream>>>(h, We, be, epre);
    hipMemsetAsync(e, 0, (size_t)B_ * O_ * sizeof(float), stream);
    k2_escore   <<<dim3(256, 4), 256, 0, stream>>>(xj, Ue, wv, epre, e);
    k3_softmax  <<<B_, 128, 0, stream>>>(e, mask, a);
    k4_phi_concat<<<B_, 256, 0, stream>>>(a, xj, XF, x);
    k5_gates    <<<dim3(4, 16, 2), 256, 0, stream>>>(x, h, c, Wi, Wf, Wc, Wo,
                                                     Ui, Uf, Uc, Uo, Vi, Vf, Vo, pre);
    k6_lstm     <<<1024, 256, 0, stream>>>(pre, c, bi, bf, bc, bo, out);
}